// SA_69741678952786
// MI455X (gfx1250) — compile-verified
//
#include <hip/hip_runtime.h>
#include <math.h>

typedef __attribute__((ext_vector_type(2))) float v2f;
typedef __attribute__((ext_vector_type(8))) float v8f;
typedef __attribute__((ext_vector_type(4))) int   v4i;

#define DIMF 192
#define HP   96
#define NPOS 9216   // 96*96
#define NC   48
#define DH   16
#define NT   (NPOS / 16)   // 576 n-tiles
#define TQ   (16 * DH)     // 256 floats / tile
#define TV   (16 * NC)     // 768 floats / tile

#if __has_builtin(__builtin_amdgcn_global_load_async_to_lds_b128)
#define HAVE_ASYNC_LDS 1
#endif

static __device__ __forceinline__ void wait_async_zero() {
#if defined(HAVE_ASYNC_LDS)
#if __has_builtin(__builtin_amdgcn_s_wait_asynccnt)
    __builtin_amdgcn_s_wait_asynccnt(0);
#else
    asm volatile("s_wait_asynccnt 0x0" ::: "memory");
#endif
#endif
}

// ---------------------------------------------------------------------------
// Kernel 1: 2x2 average pool + q/k/v projections (tiny GEMMs, one thread per n)
// x layout  : [c][n]   (c*NPOS + n)
// q layout  : [n][d]   (n*16 + d)
// k layout  : [n][d]   (n*16 + d)   (k[d, n] of the reference, row-major by n)
// v layout  : [n][c]   (n*48 + c)
// ---------------------------------------------------------------------------
__global__ void SA_pool_qkv(const float* __restrict__ in,
                            const float* __restrict__ qw, const float* __restrict__ qb,
                            const float* __restrict__ kw, const float* __restrict__ kb,
                            const float* __restrict__ vw, const float* __restrict__ vb,
                            float* __restrict__ x, float* __restrict__ qarr,
                            float* __restrict__ karr, float* __restrict__ varr) {
    int n = blockIdx.x * blockDim.x + threadIdx.x;
    if (n >= NPOS) return;
    int i = n / HP, j = n % HP;
    float xc[NC];
    const float* base = in + (size_t)(2 * i) * DIMF + 2 * j;
    #pragma unroll 4
    for (int c = 0; c < NC; ++c) {
        const float* p = base + (size_t)c * DIMF * DIMF;
        float s = p[0] + p[1] + p[DIMF] + p[DIMF + 1];
        xc[c] = 0.25f * s;
        x[(size_t)c * NPOS + n] = xc[c];
    }
    #pragma unroll
    for (int d = 0; d < DH; ++d) {
        float aq = qb[d], ak = kb[d];
        for (int c = 0; c < NC; ++c) {
            aq = fmaf(qw[d * NC + c], xc[c], aq);
            ak = fmaf(kw[d * NC + c], xc[c], ak);
        }
        qarr[(size_t)n * DH + d] = aq;
        karr[(size_t)n * DH + d] = ak;
    }
    for (int o = 0; o < NC; ++o) {
        float av = vb[o];
        for (int c = 0; c < NC; ++c) av = fmaf(vw[o * NC + c], xc[c], av);
        varr[(size_t)n * NC + o] = av;
    }
}

// ---------------------------------------------------------------------------
// Kernel 2: flash-style attention with V_WMMA_F32_16X16X4_F32 and
// double-buffered GLOBAL_LOAD_ASYNC_TO_LDS_B128 staging of the q/v n-tiles.
//   S-tile:  D[M=n(16), N=m(16)] = q-tile(A) x k-tile^T(B), K = head dim 16
//   softmax over n  -> column-wise online softmax on the S tile
//   O-tile:  D[M=c(16), N=m(16)] += v-tile(A) x P(B)
// Block = 128 threads (4 waves). Each wave owns 16 m-positions; the block
// shares the staged q/v n-tiles. One barrier per iteration.
// ---------------------------------------------------------------------------
__global__ __launch_bounds__(128)
void SA_attn(const float* __restrict__ qarr, const float* __restrict__ karr,
             const float* __restrict__ varr, const float* __restrict__ x,
             const float* __restrict__ gammap, float* __restrict__ ob) {
    __shared__ float qs[2][TQ];
    __shared__ float vs[2][TV];

    const int tid  = threadIdx.x;
    const int lane = tid & 31;
    const int wid  = tid >> 5;
    const int l16  = lane & 15;
    const int hi   = lane >> 4;          // 0: lanes 0-15, 1: lanes 16-31
    const int m_base = (blockIdx.x * 4 + wid) * 16;

    // --- tile stager: copy q tile (1KB) + v tile (3KB) into buffer `buf` ---
    auto stage = [&](int buf, int nt) {
#if defined(HAVE_ASYNC_LDS)
        const float* qg = qarr + (size_t)nt * TQ + tid * 4;
        const float* vg = varr + (size_t)nt * TV + tid * 4;
        if (tid < 64)
            __builtin_amdgcn_global_load_async_to_lds_b128(
                (v4i*)qg, (v4i*)&qs[buf][tid * 4], 0, 0);
        __builtin_amdgcn_global_load_async_to_lds_b128(
            (v4i*)vg, (v4i*)&vs[buf][tid * 4], 0, 0);
        if (tid < 64)
            __builtin_amdgcn_global_load_async_to_lds_b128(
                (v4i*)(vg + 512), (v4i*)&vs[buf][512 + tid * 4], 0, 0);
#else
        for (int t = tid; t < TQ; t += 128) qs[buf][t] = qarr[(size_t)nt * TQ + t];
        for (int t = tid; t < TV; t += 128) vs[buf][t] = varr[(size_t)nt * TV + t];
#endif
    };

    // B-fragments of k for the S product, fixed for the whole n-loop.
    // B[K=d, N=m]: vgpr0 = {K=4k+0 | K=4k+2}, vgpr1 = {K=4k+1 | K=4k+3}, N = lane%16
    v2f kfrag[4];
    #pragma unroll
    for (int k = 0; k < 4; ++k) {
        const float* p = karr + (size_t)(m_base + l16) * DH + 4 * k + 2 * hi;
        kfrag[k].x = p[0];
        kfrag[k].y = p[1];
    }

    v8f o0 = {0.f, 0.f, 0.f, 0.f, 0.f, 0.f, 0.f, 0.f};
    v8f o1 = o0, o2 = o0;
    float row_max = -3.0e38f, row_sum = 0.0f;

    // prologue: stage tile 0
    stage(0, 0);
    wait_async_zero();
    __syncthreads();

    for (int nt = 0; nt < NT; ++nt) {
        const int cur = nt & 1;
        if (nt + 1 < NT) stage(cur ^ 1, nt + 1);   // overlap next-tile DMA
        const float* qsc = qs[cur];
        const float* vsc = vs[cur];

        // ---- S = Q x K^T  (M=n, N=m, K=head dim 16 -> 4 chained WMMAs) ----
        v8f s = {0.f, 0.f, 0.f, 0.f, 0.f, 0.f, 0.f, 0.f};
        #pragma unroll
        for (int k = 0; k < 4; ++k) {
            v2f a;
            const float* p = qsc + l16 * DH + 4 * k + 2 * hi;
            a.x = p[0];
            a.y = p[1];
            s = __builtin_amdgcn_wmma_f32_16x16x4_f32(false, a, false, kfrag[k],
                                                      (short)0, s, false, false);
        }

        // ---- online softmax over n (column-wise on the tile) ----
        float tmax = fmaxf(fmaxf(fmaxf(s[0], s[1]), fmaxf(s[2], s[3])),
                           fmaxf(fmaxf(s[4], s[5]), fmaxf(s[6], s[7])));
        tmax = fmaxf(tmax, __shfl_xor(tmax, 16));
        float nmax  = fmaxf(row_max, tmax);
        float scale = __expf(row_max - nmax);
        v8f p;
        float tsum = 0.f;
        #pragma unroll
        for (int j = 0; j < 8; ++j) { p[j] = __expf(s[j] - nmax); tsum += p[j]; }
        tsum += __shfl_xor(tsum, 16);
        row_sum = row_sum * scale + tsum;
        row_max = nmax;
        #pragma unroll
        for (int j = 0; j < 8; ++j) { o0[j] *= scale; o1[j] *= scale; o2[j] *= scale; }

        // ---- convert P (C/D layout) into B fragments via half swaps ----
        float x0 = __shfl_xor(p[0], 16), x1 = __shfl_xor(p[1], 16);
        float x2 = __shfl_xor(p[2], 16), x3 = __shfl_xor(p[3], 16);
        float x4 = __shfl_xor(p[4], 16), x5 = __shfl_xor(p[5], 16);
        float x6 = __shfl_xor(p[6], 16), x7 = __shfl_xor(p[7], 16);
        v2f pb[4];
        pb[0].x = hi ? x2   : p[0];  pb[0].y = hi ? x3   : p[1];  // n = 0..3
        pb[1].x = hi ? x6   : p[4];  pb[1].y = hi ? x7   : p[5];  // n = 4..7
        pb[2].x = hi ? p[2] : x0;    pb[2].y = hi ? p[3] : x1;    // n = 8..11
        pb[3].x = hi ? p[6] : x4;    pb[3].y = hi ? p[7] : x5;    // n = 12..15

        // ---- O += V x P   (A: M=c row = lane%16, K=n) ----
        #pragma unroll
        for (int k = 0; k < 4; ++k) {
            const int na = 4 * k + 2 * hi;
            v2f a0, a1, a2;
            a0.x = vsc[na * NC + 0  + l16];  a0.y = vsc[(na + 1) * NC + 0  + l16];
            a1.x = vsc[na * NC + 16 + l16];  a1.y = vsc[(na + 1) * NC + 16 + l16];
            a2.x = vsc[na * NC + 32 + l16];  a2.y = vsc[(na + 1) * NC + 32 + l16];
            o0 = __builtin_amdgcn_wmma_f32_16x16x4_f32(false, a0, false, pb[k],
                                                       (short)0, o0, false, false);
            o1 = __builtin_amdgcn_wmma_f32_16x16x4_f32(false, a1, false, pb[k],
                                                       (short)0, o1, false, false);
            o2 = __builtin_amdgcn_wmma_f32_16x16x4_f32(false, a2, false, pb[k],
                                                       (short)0, o2, false, false);
        }

        // next-tile DMA must be complete in all waves before buffers swap
        wait_async_zero();
        __syncthreads();
    }

    // ---- finalize: out[c,m] = gamma * o/row_sum + x[c,m] ----
    const float inv = gammap[0] / row_sum;   // identical in both lane halves
    const int m = m_base + l16;
    #pragma unroll
    for (int j = 0; j < 8; ++j) {
        const int c0 = j + 8 * hi;
        ob[(size_t)(c0)      * NPOS + m] = o0[j] * inv + x[(size_t)(c0)      * NPOS + m];
        ob[(size_t)(c0 + 16) * NPOS + m] = o1[j] * inv + x[(size_t)(c0 + 16) * NPOS + m];
        ob[(size_t)(c0 + 32) * NPOS + m] = o2[j] * inv + x[(size_t)(c0 + 32) * NPOS + m];
    }
}

// ---------------------------------------------------------------------------
// Kernel 3: ConvTranspose2d(48->48, k=4, s=2, p=1).
// y[co,oy,ox] = db[co] + sum_ci sum_{ky,kx valid} ob[ci,iy,ix] * dw[ci,co,ky,kx]
// with iy = (oy+1-ky)/2 (parity-selected), likewise ix. One thread per output
// pixel for a fixed co; the co's 48x4x4 weight slice is cached in LDS.
// ---------------------------------------------------------------------------
__global__ __launch_bounds__(256)
void SA_convt(const float* __restrict__ ob, const float* __restrict__ dw,
              const float* __restrict__ db, float* __restrict__ y) {
    __shared__ float wsl[NC * 16];         // [ci][ky*4+kx] for this co
    const int co = blockIdx.y;
    for (int t = threadIdx.x; t < NC * 16; t += blockDim.x)
        wsl[t] = dw[((size_t)(t >> 4) * NC + co) * 16 + (t & 15)];
    __syncthreads();

    const int pix = blockIdx.x * blockDim.x + threadIdx.x;
    if (pix >= DIMF * DIMF) return;
    const int oy = pix / DIMF, ox = pix % DIMF;

    float acc = db[co];
    const int ky0 = (oy + 1) & 1;
    const int kx0 = (ox + 1) & 1;
    #pragma unroll
    for (int dyi = 0; dyi < 2; ++dyi) {
        const int ky = ky0 + 2 * dyi;
        const int iy = (oy + 1 - ky) >> 1;
        if ((unsigned)iy >= (unsigned)HP) continue;
        #pragma unroll
        for (int dxi = 0; dxi < 2; ++dxi) {
            const int kx = kx0 + 2 * dxi;
            const int ix = (ox + 1 - kx) >> 1;
            if ((unsigned)ix >= (unsigned)HP) continue;
            const float* xp = ob + (size_t)iy * HP + ix;
            const float* wp = wsl + ky * 4 + kx;
            #pragma unroll 8
            for (int ci = 0; ci < NC; ++ci)
                acc = fmaf(xp[(size_t)ci * NPOS], wp[ci * 16], acc);
        }
    }
    y[(size_t)co * DIMF * DIMF + pix] = acc;
}

// ---------------------------------------------------------------------------
extern "C" void kernel_launch(void* const* d_in, const int* in_sizes, int n_in,
                              void* d_out, int out_size, void* d_ws, size_t ws_size,
                              hipStream_t stream) {
    const float* in    = (const float*)d_in[0];
    const float* qw    = (const float*)d_in[1];
    const float* qb    = (const float*)d_in[2];
    const float* kw    = (const float*)d_in[3];
    const float* kb    = (const float*)d_in[4];
    const float* vw    = (const float*)d_in[5];
    const float* vb    = (const float*)d_in[6];
    const float* gamma = (const float*)d_in[7];
    const float* dw    = (const float*)d_in[8];
    const float* db    = (const float*)d_in[9];
    float* y = (float*)d_out;

    float* ws   = (float*)d_ws;
    float* x    = ws;                       // 48*9216   = 442368
    float* qarr = x    + (size_t)NC * NPOS; // 9216*16   = 147456
    float* karr = qarr + (size_t)NPOS * DH; // 9216*16   = 147456
    float* varr = karr + (size_t)NPOS * DH; // 9216*48   = 442368
    float* ob   = varr + (size_t)NPOS * NC; // 48*9216   = 442368
    (void)in_sizes; (void)n_in; (void)out_size; (void)ws_size;

    SA_pool_qkv<<<(NPOS + 255) / 256, 256, 0, stream>>>(
        in, qw, qb, kw, kb, vw, vb, x, qarr, karr, varr);

    SA_attn<<<NPOS / 64, 128, 0, stream>>>(qarr, karr, varr, x, gamma, ob);

    dim3 cgrid((DIMF * DIMF + 255) / 256, NC);
    SA_convt<<<cgrid, 256, 0, stream>>>(ob, dw, db, y);
}